// Loss_89730456748593
// MI455X (gfx1250) — compile-verified
//
#include <hip/hip_runtime.h>
#include <hip/hip_bf16.h>
#include <math.h>

// Problem constants (from reference)
#define S_ 8
#define B_ 2048
#define C_ 20
#define INV_BW 2.5f  // 1 / 0.4 Laplacian bandwidth

typedef __attribute__((ext_vector_type(2))) float v2f;
typedef __attribute__((ext_vector_type(8))) float v8f;

__device__ __forceinline__ float wave_reduce_add(float v) {
#pragma unroll
  for (int off = 16; off > 0; off >>= 1) v += __shfl_xor(v, off, 32);
  return v;
}

// ---------------------------------------------------------------------------
// Kernel 1: per-row softmax stats. idx = s*B + b.
//   p    = max softmax prob = 1 / sum(exp(l - max))
//   acc  = (argmax == label)
//   ce   = -log_softmax[label] = max + log(sumexp) - l[label]
// Deterministic block reductions -> ce_part[64], acc_part[64].
// ---------------------------------------------------------------------------
__global__ __launch_bounds__(256) void k_stats(const float* __restrict__ logits,
                                               const int* __restrict__ labels,
                                               float* __restrict__ p_arr,
                                               float* __restrict__ acc_arr,
                                               float* __restrict__ ce_part,
                                               float* __restrict__ acc_part) {
  int idx = blockIdx.x * 256 + threadIdx.x;  // [0, S*B)
  int b = idx & (B_ - 1);

  float l[C_];
  const float4* lp = (const float4*)(logits + (size_t)idx * C_);  // 80B row, 16B aligned
#pragma unroll
  for (int q = 0; q < 5; ++q) {
    float4 v = lp[q];
    l[q * 4 + 0] = v.x; l[q * 4 + 1] = v.y; l[q * 4 + 2] = v.z; l[q * 4 + 3] = v.w;
  }

  float m = l[0];
  int am = 0;
#pragma unroll
  for (int k = 1; k < C_; ++k)
    if (l[k] > m) { m = l[k]; am = k; }  // strict > == first-occurrence argmax

  float se = 0.f;
#pragma unroll
  for (int k = 0; k < C_; ++k) se += expf(l[k] - m);

  int lab = labels[b];
  float ce = m + logf(se) - l[lab];
  float acc = (am == lab) ? 1.f : 0.f;

  p_arr[idx] = 1.f / se;
  acc_arr[idx] = acc;

  float cw = wave_reduce_add(ce);
  float aw = wave_reduce_add(acc);
  __shared__ float sce[8], sac[8];
  int lane = threadIdx.x & 31, wv = threadIdx.x >> 5;
  if (lane == 0) { sce[wv] = cw; sac[wv] = aw; }
  __syncthreads();
  if (threadIdx.x == 0) {
    float c = 0.f, a = 0.f;
#pragma unroll
    for (int k = 0; k < 8; ++k) { c += sce[k]; a += sac[k]; }
    ce_part[blockIdx.x] = c;
    acc_part[blockIdx.x] = a;
  }
}

// ---------------------------------------------------------------------------
// Kernel 2: fixed-order reduce of stats -> ncorrect[s], ce_sum.
// (k_stats blocks: 8 per sample, sample s owns blocks [s*8, s*8+8))
// ---------------------------------------------------------------------------
__global__ void k_redstats(const float* __restrict__ ce_part,
                           const float* __restrict__ acc_part,
                           float* __restrict__ ncorr,
                           float* __restrict__ ce_sum) {
  int t = threadIdx.x;
  if (t < S_) {
    float s = 0.f;
#pragma unroll
    for (int k = 0; k < 8; ++k) s += acc_part[t * 8 + k];
    ncorr[t] = s;
  }
  if (t == S_) {
    float s = 0.f;
    for (int k = 0; k < 64; ++k) s += ce_part[k];
    ce_sum[0] = s;
  }
}

// ---------------------------------------------------------------------------
// Kernel 3: per-row MMCE weights + exp factors.
//   w  = (acc - p) * (acc==1 ? 1/B : 1/(ncorrect - B))   (0 if denom==0)
//   em = exp(-p/0.4), ep = exp(+p/0.4)   (p in [0.05,1] -> no overflow)
// exp(-|p_i-p_j|/0.4) == min(em_i*ep_j, ep_i*em_j)  (exact, ties ok)
// ---------------------------------------------------------------------------
__global__ __launch_bounds__(256) void k_prep(const float* __restrict__ p_arr,
                                              const float* __restrict__ acc_arr,
                                              const float* __restrict__ ncorr,
                                              float* __restrict__ em,
                                              float* __restrict__ ep,
                                              float* __restrict__ w) {
  int idx = blockIdx.x * 256 + threadIdx.x;
  int s = idx >> 11;
  float nc = ncorr[s];
  float denom = nc - (float)B_;
  float rinc = (denom != 0.f) ? (1.f / denom) : 0.f;
  float p = p_arr[idx], a = acc_arr[idx];
  float wv = (a - p) * ((a == 1.f) ? (1.f / (float)B_) : rinc);
  w[idx] = wv;
  em[idx] = expf(-INV_BW * p);
  ep[idx] = expf(INV_BW * p);
}

// ---------------------------------------------------------------------------
// Kernel 4: pairwise quadratic form via WMMA rank-1 outer products.
// Grid: 128 blocks (s = bx>>4, chunk = bx&15), 256 threads = 8 waves.
// Each wave owns a 16-row i-tile; streams all 2048 j in 16-wide steps.
// Per step: two v_wmma_f32_16x16x4_f32 compute em_i*ep_j and ep_i*em_j
// (K=0 column only => exact outer products), VALU does min * w_j, accumulate.
// Inputs staged into LDS with GLOBAL_LOAD_ASYNC_TO_LDS_B128 (ASYNCcnt path).
// Operand builds are branchless (unconditional LDS broadcast load * lane
// mask) and software-pipelined so ds latency overlaps WMMA+VALU work.
// ---------------------------------------------------------------------------
__global__ __launch_bounds__(256) void k_pairwise(const float* __restrict__ em_g,
                                                  const float* __restrict__ ep_g,
                                                  const float* __restrict__ w_g,
                                                  float* __restrict__ pair_part) {
  __shared__ __align__(16) float s_em[B_];
  __shared__ __align__(16) float s_ep[B_];
  __shared__ __align__(16) float s_w[B_];

  const int tid = threadIdx.x;
  const int s = blockIdx.x >> 4;
  const int chunk = blockIdx.x & 15;
  const float* gem = em_g + (size_t)s * B_;
  const float* gep = ep_g + (size_t)s * B_;
  const float* gw  = w_g  + (size_t)s * B_;

  // Async staging: 24KB (3 x 2048 floats) via b128 copies on ASYNCcnt.
  for (int k = tid * 4; k < B_; k += 256 * 4) {
    unsigned int l0 = (unsigned int)(size_t)(&s_em[k]);
    unsigned int l1 = (unsigned int)(size_t)(&s_ep[k]);
    unsigned int l2 = (unsigned int)(size_t)(&s_w[k]);
    unsigned long long g0 = (unsigned long long)(size_t)(gem + k);
    unsigned long long g1 = (unsigned long long)(size_t)(gep + k);
    unsigned long long g2 = (unsigned long long)(size_t)(gw + k);
    asm volatile("global_load_async_to_lds_b128 %0, %1, off" ::"v"(l0), "v"(g0) : "memory");
    asm volatile("global_load_async_to_lds_b128 %0, %1, off" ::"v"(l1), "v"(g1) : "memory");
    asm volatile("global_load_async_to_lds_b128 %0, %1, off" ::"v"(l2), "v"(g2) : "memory");
  }
  asm volatile("s_wait_asynccnt 0" ::: "memory");
  __syncthreads();

  const int wave = tid >> 5;
  const int lane = tid & 31;
  const int hi = lane >> 4;       // 0: rows g, 1: rows g+8 (C/D layout)
  const int col = lane & 15;      // D column / broadcast index for this lane
  const int ibase = (chunk * 8 + wave) * 16;
  const float msk = (lane < 16) ? 1.f : 0.f;  // K=0 lives in lanes 0-15

  // w for the 8 rows this lane's accumulator VGPRs represent
  float wrow[8];
#pragma unroll
  for (int g = 0; g < 8; ++g) wrow[g] = s_w[ibase + g + 8 * hi];

  // A-matrix (16x4 f32): VGPR0 lanes0-15 = K=0 row values (M=lane); rest zero.
  v2f a_em, a_ep;
  a_em[0] = s_em[ibase + col] * msk; a_em[1] = 0.f;
  a_ep[0] = s_ep[ibase + col] * msk; a_ep[1] = 0.f;

  v8f acc = {};
  // software pipeline: j-step operands loaded one iteration ahead
  float vep = s_ep[col];
  float vem = s_em[col];
  float vw  = s_w[col];
#pragma unroll 2
  for (int jb = 0; jb < B_; jb += 16) {
    const int jn = (jb + 16) & (B_ - 1);  // wraps to 0 on last iter (unused)
    float nep = s_ep[jn + col];
    float nem = s_em[jn + col];
    float nw  = s_w[jn + col];

    // B-matrix (4x16 f32): VGPR0 lanes0-15 = K=0 row (N=lane); rest zero.
    v2f b_ep, b_em;
    b_ep[0] = vep * msk; b_ep[1] = 0.f;
    b_em[0] = vem * msk; b_em[1] = 0.f;

    v8f z = {};
    // t1[i][j] = em_i * ep_j ; t2[i][j] = ep_i * em_j  (exact rank-1 products)
    v8f t1 = __builtin_amdgcn_wmma_f32_16x16x4_f32(false, a_em, false, b_ep,
                                                   (short)0, z, false, false);
    v8f t2 = __builtin_amdgcn_wmma_f32_16x16x4_f32(false, a_ep, false, b_em,
                                                   (short)0, z, false, false);
#pragma unroll
    for (int g = 0; g < 8; ++g)
      acc[g] = fmaf(fminf(t1[g], t2[g]), vw, acc[g]);  // K_ij * w_j

    vep = nep; vem = nem; vw = nw;
  }

  // dot accumulators with w_i, then wave-sum covers cols + both halves
  float part = 0.f;
#pragma unroll
  for (int g = 0; g < 8; ++g) part = fmaf(acc[g], wrow[g], part);
  part = wave_reduce_add(part);
  if (lane == 0) pair_part[blockIdx.x * 8 + wave] = part;  // sample s: [s*128,(s+1)*128)
}

// ---------------------------------------------------------------------------
// Kernel 5: finalize. loss = 2 * mean_s sqrt(sum_s / B^2) + ce_sum/(S*B)
// ---------------------------------------------------------------------------
__global__ void k_final(const float* __restrict__ pair_part,
                        const float* __restrict__ ce_sum,
                        float* __restrict__ out) {
  __shared__ float sm[S_];
  int t = threadIdx.x;
  if (t < S_) {
    float s = 0.f;
    for (int k = 0; k < 128; ++k) s += pair_part[t * 128 + k];
    sm[t] = sqrtf(fmaxf(s, 0.f) * (1.f / ((float)B_ * (float)B_)));
  }
  __syncthreads();
  if (t == 0) {
    float mm = 0.f;
#pragma unroll
    for (int i = 0; i < S_; ++i) mm += sm[i];
    mm *= (1.f / (float)S_);
    out[0] = 2.0f * mm + ce_sum[0] * (1.f / (float)(S_ * B_));
  }
}

// ---------------------------------------------------------------------------
extern "C" void kernel_launch(void* const* d_in, const int* in_sizes, int n_in,
                              void* d_out, int out_size, void* d_ws, size_t ws_size,
                              hipStream_t stream) {
  (void)in_sizes; (void)n_in; (void)out_size; (void)ws_size;
  const float* logits = (const float*)d_in[0];  // [S,B,C] f32
  const int* labels = (const int*)d_in[1];      // [B] i32
  float* ws = (float*)d_ws;

  const int SB = S_ * B_;  // 16384
  float* p_arr    = ws;             // 16384
  float* acc_arr  = ws + SB;        // 16384
  float* em       = ws + 2 * SB;    // 16384
  float* ep       = ws + 3 * SB;    // 16384
  float* w        = ws + 4 * SB;    // 16384
  float* ce_part  = ws + 5 * SB;         // 64
  float* acc_part = ws + 5 * SB + 64;    // 64
  float* ncorr    = ws + 5 * SB + 128;   // 8
  float* ce_sum   = ws + 5 * SB + 136;   // 1
  float* pair_part= ws + 5 * SB + 144;   // 1024

  k_stats<<<SB / 256, 256, 0, stream>>>(logits, labels, p_arr, acc_arr, ce_part, acc_part);
  k_redstats<<<1, 16, 0, stream>>>(ce_part, acc_part, ncorr, ce_sum);
  k_prep<<<SB / 256, 256, 0, stream>>>(p_arr, acc_arr, ncorr, em, ep, w);
  k_pairwise<<<S_ * 16, 256, 0, stream>>>(em, ep, w, pair_part);
  k_final<<<1, 8, 0, stream>>>(pair_part, ce_sum, (float*)d_out);
}